// StateSpaceLayer_1589137900158
// MI455X (gfx1250) — compile-verified
//
#include <hip/hip_runtime.h>

typedef __attribute__((ext_vector_type(2))) float v2f;
typedef __attribute__((ext_vector_type(8))) float v8f;

#define SEQ 512
#define D   64
#define T   16
#define NCHUNK (SEQ / T)

// out[b,t,r,c] = sum_{j<=t} A_r^(t-j) x[b,j,r,c]
// Chunked scan: within-chunk 16x16 decay matmul via V_WMMA_F32_16X16X4_F32,
// cross-chunk carry via a 32-step recurrence per wave.
__global__ __launch_bounds__(128) void ssm_chunked_scan_wmma(
    const float* __restrict__ x, const float* __restrict__ Avec,
    float* __restrict__ out)
{
  const int  lane = threadIdx.x & 31;
  const int  wv   = threadIdx.x >> 5;      // c-tile 0..3
  const int  b    = blockIdx.x >> 6;       // 0..3
  const int  r    = blockIdx.x & 63;       // 0..63
  const int  c0   = wv * 16;
  const int  n    = lane & 15;             // col within tile; also row M for A-frag
  const bool hi   = (lane >= 16);
  const int  c    = c0 + n;

  const float a = fmaxf(Avec[r], 1e-6f);   // clip like the reference

  // power table a^0 .. a^15, and a^16
  float p[16];
  p[0] = 1.0f;
  #pragma unroll
  for (int i = 1; i < 16; ++i) p[i] = p[i-1] * a;
  const float a16 = p[15] * a;

  // A-matrix fragments: M[t][j] = (t>=j) ? a^(t-j) : 0   (t = n, per 16x4 layout)
  // lanes 0-15 supply K columns {4k,4k+1}; lanes 16-31 supply {4k+2,4k+3}.
  const int jo = hi ? 2 : 0;
  v2f afrag[4];
  #pragma unroll
  for (int k = 0; k < 4; ++k) {
    const int j0 = 4*k + jo;
    afrag[k].x = (n >= j0    ) ? p[n - j0    ] : 0.0f;
    afrag[k].y = (n >= j0 + 1) ? p[n - j0 - 1] : 0.0f;
  }

  // carry-decay per D VGPR v: a^(t_local+1), t_local = v + (hi?8:0)
  float dpow[8];
  #pragma unroll
  for (int v = 0; v < 8; ++v) {
    const int e = v + (hi ? 8 : 0) + 1;    // 1..16
    dpow[v] = (e == 16) ? a16 : p[e];
  }

  // element (b, j, r, c) lives at ((b*SEQ + j)*D + r)*D + c
  const float* xb = x   + ((size_t)b * SEQ * D + r) * D;
  float*       ob = out + ((size_t)b * SEQ * D + r) * D;

  float state = 0.0f;   // carry S for column c (duplicated in lanes n and n+16)

  for (int chunk = 0; chunk < NCHUNK; ++chunk) {
    const int jbase = chunk * T;

    // prefetch next chunk (one 64B row segment per lane -> global_prefetch_b8)
    if (chunk + 1 < NCHUNK) {
      const float* pf = xb + (size_t)(jbase + T + n) * D * D + c0;
      __builtin_prefetch(pf, 0, 1);
    }

    // B fragments: X[j][c] with B 4x16 layout mirroring A's lane split
    v2f bfrag[4];
    #pragma unroll
    for (int k = 0; k < 4; ++k) {
      const int j0 = jbase + 4*k + jo;
      bfrag[k].x = xb[(size_t) j0      * D * D + c];
      bfrag[k].y = xb[(size_t)(j0 + 1) * D * D + c];
    }

    // local 16x16 result = M x X  (K=16 via four K=4 WMMAs)
    v8f cf = {0.f,0.f,0.f,0.f,0.f,0.f,0.f,0.f};
    #pragma unroll
    for (int k = 0; k < 4; ++k) {
      cf = __builtin_amdgcn_wmma_f32_16x16x4_f32(
          false, afrag[k], false, bfrag[k], (short)0, cf, false, false);
    }

    // out[t] = local[t] + a^(t_local+1) * S_old
    #pragma unroll
    for (int v = 0; v < 8; ++v) {
      const int t = jbase + v + (hi ? 8 : 0);
      ob[(size_t)t * D * D + c] = cf[v] + dpow[v] * state;
    }

    // S = a^16 * S + local_row15 ; row15 is VGPR7 of lanes 16-31
    const int srcByte = (n + 16) << 2;
    const float row15 = __int_as_float(
        __builtin_amdgcn_ds_bpermute(srcByte, __float_as_int(cf[7])));
    state = a16 * state + row15;
  }
}

extern "C" void kernel_launch(void* const* d_in, const int* in_sizes, int n_in,
                              void* d_out, int out_size, void* d_ws, size_t ws_size,
                              hipStream_t stream) {
  (void)in_sizes; (void)n_in; (void)d_ws; (void)ws_size; (void)out_size;
  const float* x = (const float*)d_in[0];
  const float* A = (const float*)d_in[1];
  float* out = (float*)d_out;

  dim3 grid(4 * 64);   // (b, r)
  dim3 block(128);     // 4 waves, one 16-wide c-tile each
  hipLaunchKernelGGL(ssm_chunked_scan_wmma, grid, block, 0, stream, x, A, out);
}